// QFCModel_65481071403453
// MI455X (gfx1250) — compile-verified
//
#include <hip/hip_runtime.h>
#include <hip/hip_bf16.h>

#define BATCH   65536
#define BN_EPS  1e-5f

typedef __attribute__((ext_vector_type(2))) float v2f;
typedef __attribute__((ext_vector_type(4))) float v4f;
typedef __attribute__((ext_vector_type(8))) float v8f;

__device__ __forceinline__ float2 cmul(float2 a, float2 b) {
    return make_float2(a.x * b.x - a.y * b.y, a.x * b.y + a.y * b.x);
}
__device__ __forceinline__ float2 cadd(float2 a, float2 b) {
    return make_float2(a.x + b.x, a.y + b.y);
}

// ---------------------------------------------------------------------------
// Kernel 0: precompute fused RZ@RX gates M[l][w] (2x2 complex each) and zero
// the batchnorm statistics accumulators. 12 gate threads + 8 zeroing threads.
// M[l][w] with c0=cos(t0/2), s0=sin(t0/2), c1=cos(t1/2), s1=sin(t1/2):
//   M00 = ( c1*c0, -c1*s0)   M01 = ( s1*s0, -s1*c0)
//   M10 = (-s1*s0, -s1*c0)   M11 = ( c1*c0,  c1*s0)
// ---------------------------------------------------------------------------
__global__ void prep_kernel(const float* __restrict__ params,
                            float* __restrict__ Mg,
                            float* __restrict__ stats) {
    const int t = threadIdx.x;
    if (t < 8) stats[t] = 0.0f;
    if (t < 12) {
        const float t0 = params[t * 3 + 0];
        const float t1 = params[t * 3 + 1];
        float c0, s0, c1, s1;
        __sincosf(0.5f * t0, &s0, &c0);
        __sincosf(0.5f * t1, &s1, &c1);
        float2* Mp = (float2*)Mg + t * 4;
        Mp[0] = make_float2( c1 * c0, -c1 * s0);
        Mp[1] = make_float2( s1 * s0, -s1 * c0);
        Mp[2] = make_float2(-s1 * s0, -s1 * c0);
        Mp[3] = make_float2( c1 * c0,  c1 * s0);
    }
}

// ---------------------------------------------------------------------------
// Kernel 1: bandwidth-bound chunk-mean reduction (the HBM roofline phase:
// 205 MB @ 23.3 TB/s ~ 9 us). One wave32 per sample; 196 float4 per sample
// streamed with coalesced non-temporal b128 loads (x is a one-shot 205 MB
// stream > 192 MB L2 -- NT keeps it from evicting enc, which kernel 2
// re-reads from L2). 4 segmented accumulators reduced with a shfl_xor tree.
// ---------------------------------------------------------------------------
__global__ __launch_bounds__(256) void enc_kernel(const float* __restrict__ x,
                                                  float* __restrict__ enc) {
    const int lane   = threadIdx.x & 31;
    const int wid    = threadIdx.x >> 5;
    const int sample = blockIdx.x * 8 + wid;

    const v4f* __restrict__ xs = (const v4f*)x + (size_t)sample * 196;

    // prefetch next sample block's lines (gfx1250 global_prefetch_b8)
    __builtin_prefetch((const char*)(xs + 196) + lane * 128, 0, 0);

    float a0 = 0.f, a1 = 0.f, a2 = 0.f, a3 = 0.f;
#pragma unroll
    for (int i = 0; i < 6; ++i) {
        const int f = lane + 32 * i;            // float4 index, 0..191
        const v4f v = __builtin_nontemporal_load(&xs[f]);
        const float s = (v.x + v.y) + (v.z + v.w);
        if (f < 49)       a0 += s;              // chunk boundaries: 49/98/147
        else if (f < 98)  a1 += s;
        else if (f < 147) a2 += s;
        else              a3 += s;
    }
    if (lane < 4) {                             // remainder f=192..195, chunk 3
        const v4f v = __builtin_nontemporal_load(&xs[192 + lane]);
        a3 += (v.x + v.y) + (v.z + v.w);
    }

#pragma unroll
    for (int m = 16; m >= 1; m >>= 1) {
        a0 += __shfl_xor(a0, m, 32);
        a1 += __shfl_xor(a1, m, 32);
        a2 += __shfl_xor(a2, m, 32);
        a3 += __shfl_xor(a3, m, 32);
    }
    if (lane == 0) {
        const float inv = 1.0f / 196.0f;
        v4f e;
        e.x = a0 * inv; e.y = a1 * inv; e.z = a2 * inv; e.w = a3 * inv;
        ((v4f*)enc)[sample] = e;
    }
}

// ---------------------------------------------------------------------------
// Kernel 2: per-lane 4-qubit statevector sim (16 complex amps in registers,
// fully unrolled -> all qubit bit masks constant-folded), then the Linear(4,4)
// done with V_WMMA_F32_16X16X4_F32 (exact f32): each wave's 32 samples feed
// two WMMAs (A = 16x4 z-block, B = W^T in 4 of 16 columns, C = bias).
// BN statistics accumulated via LDS block reduction + 8 atomics per block.
// ---------------------------------------------------------------------------
__global__ __launch_bounds__(256) void sim_kernel(const float* __restrict__ enc,
                                                  const float* __restrict__ Mg,
                                                  const float* __restrict__ W4,
                                                  const float* __restrict__ bias,
                                                  float* __restrict__ outp,
                                                  float* __restrict__ stats) {
    const int tid  = blockIdx.x * blockDim.x + threadIdx.x;   // sample index
    const int lane = threadIdx.x & 31;

    // ---- encoding angles -> RY(c,s) per wire ----
    const v4f ev = ((const v4f*)enc)[tid];
    float cw[4], sw[4];
    __sincosf(0.5f * ev.x, &sw[0], &cw[0]);
    __sincosf(0.5f * ev.y, &sw[1], &cw[1]);
    __sincosf(0.5f * ev.z, &sw[2], &cw[2]);
    __sincosf(0.5f * ev.w, &sw[3], &cw[3]);

    // ---- statevector |0000> ----
    float2 st[16];
#pragma unroll
    for (int i = 0; i < 16; ++i) st[i] = make_float2(0.f, 0.f);
    st[0] = make_float2(1.f, 0.f);

#pragma unroll
    for (int l = 0; l < 3; ++l) {
#pragma unroll
        for (int w = 0; w < 4; ++w) {
            const float2* Mp = (const float2*)Mg + (l * 4 + w) * 4;
            const float2 M00 = Mp[0], M01 = Mp[1], M10 = Mp[2], M11 = Mp[3];
            const float c = cw[w], s = sw[w];
            // U = M @ RY : col0 = M(:,0)*c + M(:,1)*s ; col1 = -M(:,0)*s + M(:,1)*c
            const float2 U00 = make_float2( M00.x * c + M01.x * s,  M00.y * c + M01.y * s);
            const float2 U01 = make_float2(-M00.x * s + M01.x * c, -M00.y * s + M01.y * c);
            const float2 U10 = make_float2( M10.x * c + M11.x * s,  M10.y * c + M11.y * s);
            const float2 U11 = make_float2(-M10.x * s + M11.x * c, -M10.y * s + M11.y * c);

            const int bit = 8 >> w;   // wire w is axis w of (2,2,2,2), C-order
#pragma unroll
            for (int i = 0; i < 16; ++i) {
                if (i & bit) continue;
                const float2 a = st[i], b = st[i | bit];
                st[i]       = cadd(cmul(U00, a), cmul(U01, b));
                st[i | bit] = cadd(cmul(U10, a), cmul(U11, b));
            }
            if (w < 3) {              // CNOT(control=w, target=w+1)
                const int bc = 8 >> w, bt = bc >> 1;
#pragma unroll
                for (int i = 0; i < 16; ++i) {
                    if ((i & bc) && !(i & bt)) {
                        const float2 t = st[i];
                        st[i] = st[i | bt];
                        st[i | bt] = t;
                    }
                }
            }
        }
    }

    // ---- PauliZ expvals ----
    float z[4] = {0.f, 0.f, 0.f, 0.f};
#pragma unroll
    for (int i = 0; i < 16; ++i) {
        const float p = st[i].x * st[i].x + st[i].y * st[i].y;
#pragma unroll
        for (int w = 0; w < 4; ++w) z[w] += (i & (8 >> w)) ? -p : p;
    }

    // ---- Linear(4,4) via V_WMMA_F32_16X16X4_F32, two 16-sample groups ----
    const int half = lane >> 4;        // 0: lanes 0-15 (K=0,1) ; 1: lanes 16-31 (K=2,3)
    const int nl   = lane & 15;        // B/C/D column index (features live in n<4)

    // B[k][n] = W[n][k]  (out = z @ W^T)
    float bw0 = 0.f, bw1 = 0.f;
    if (nl < 4) {
        bw0 = W4[nl * 4 + half * 2 + 0];
        bw1 = W4[nl * 4 + half * 2 + 1];
    }
    v2f B; B[0] = bw0; B[1] = bw1;

    const float cb = (nl < 4) ? bias[nl] : 0.f;   // C[m][n] = b[n]
    v8f C;
#pragma unroll
    for (int v = 0; v < 8; ++v) C[v] = cb;

    // A layout (f32 16x4): lanes 0-15 row=lane K=0/1; lanes 16-31 row=lane-16 K=2/3
    v2f A0, A1;
    {
        int src = nl;                                  // group 0: samples = lanes 0..15
        float t0 = __shfl(z[0], src, 32), t1 = __shfl(z[1], src, 32);
        float t2 = __shfl(z[2], src, 32), t3 = __shfl(z[3], src, 32);
        A0[0] = half ? t2 : t0;
        A0[1] = half ? t3 : t1;
        src = nl + 16;                                 // group 1: samples = lanes 16..31
        float u0 = __shfl(z[0], src, 32), u1 = __shfl(z[1], src, 32);
        float u2 = __shfl(z[2], src, 32), u3 = __shfl(z[3], src, 32);
        A1[0] = half ? u2 : u0;
        A1[1] = half ? u3 : u1;
    }

    v8f D0 = __builtin_amdgcn_wmma_f32_16x16x4_f32(false, A0, false, B, (short)0, C, false, false);
    v8f D1 = __builtin_amdgcn_wmma_f32_16x16x4_f32(false, A1, false, B, (short)0, C, false, false);

    // ---- store pre-BN outputs; D layout: VGPR v -> row v + 8*half, col = nl ----
    const int wavebase = blockIdx.x * blockDim.x + (threadIdx.x & ~31);
    float sm = 0.f, sq = 0.f;
    if (nl < 4) {
#pragma unroll
        for (int v = 0; v < 8; ++v) {
            const int m = v + half * 8;
            const float d0 = D0[v], d1 = D1[v];
            outp[(wavebase + m) * 4 + nl]      = d0;
            outp[(wavebase + 16 + m) * 4 + nl] = d1;
            sm += d0 + d1;
            sq += d0 * d0 + d1 * d1;
        }
    }
    // fold the half=1 partials (same feature nl) into lanes 0..3
    sm += __shfl_xor(sm, 16, 32);
    sq += __shfl_xor(sq, 16, 32);

    __shared__ float red[8][8];
    const int wid = threadIdx.x >> 5;
    if (lane < 4) { red[wid][lane] = sm; red[wid][4 + lane] = sq; }
    __syncthreads();
    if (threadIdx.x < 8) {
        float acc = 0.f;
#pragma unroll
        for (int w = 0; w < 8; ++w) acc += red[w][threadIdx.x];
        atomicAdd(&stats[threadIdx.x], acc);   // [0..3]=sum, [4..7]=sumsq
    }
}

// ---------------------------------------------------------------------------
// Kernel 3: BatchNorm1d finalize (training mode, biased variance), in place.
// ---------------------------------------------------------------------------
__global__ __launch_bounds__(256) void bn_kernel(float* __restrict__ outp,
                                                 const float* __restrict__ stats,
                                                 const float* __restrict__ gamma,
                                                 const float* __restrict__ beta) {
    const int i = blockIdx.x * blockDim.x + threadIdx.x;   // over BATCH*4
    const int n = i & 3;
    const float invB = 1.0f / (float)BATCH;
    const float mu  = stats[n] * invB;
    const float var = stats[4 + n] * invB - mu * mu;
    const float sc  = gamma[n] * rsqrtf(var + BN_EPS);
    outp[i] = (outp[i] - mu) * sc + beta[n];
}

// ---------------------------------------------------------------------------
extern "C" void kernel_launch(void* const* d_in, const int* in_sizes, int n_in,
                              void* d_out, int out_size, void* d_ws, size_t ws_size,
                              hipStream_t stream) {
    (void)in_sizes; (void)n_in; (void)out_size; (void)ws_size;

    const float* x      = (const float*)d_in[0];   // (65536,1,28,28)
    const float* params = (const float*)d_in[1];   // (3,4,3)
    const float* W      = (const float*)d_in[2];   // (4,4)
    const float* b      = (const float*)d_in[3];   // (4,)
    const float* gamma  = (const float*)d_in[4];   // (4,)
    const float* beta   = (const float*)d_in[5];   // (4,)
    float* out = (float*)d_out;                    // (65536,4)

    // workspace layout
    float* enc   = (float*)d_ws;                       // BATCH*4 floats (1 MB)
    float* Mg    = enc + (size_t)BATCH * 4;            // 48 floats (12 gates x 2x2 cplx)
    float* stats = Mg + 48;                            // 8 floats (sum[4], sumsq[4])

    prep_kernel<<<1, 32, 0, stream>>>(params, Mg, stats);
    enc_kernel<<<BATCH / 8, 256, 0, stream>>>(x, enc);
    sim_kernel<<<BATCH / 256, 256, 0, stream>>>(enc, Mg, W, b, out, stats);
    bn_kernel<<<(BATCH * 4) / 256, 256, 0, stream>>>(out, stats, gamma, beta);
}